// EllipseLoss_3582002725080
// MI455X (gfx1250) — compile-verified
//
#include <hip/hip_runtime.h>
#include <hip/hip_bf16.h>
#include <math.h>

#ifndef INFF
#define INFF __builtin_inff()
#endif

#define MAX_TGT 512

#if defined(__AMDGCN__) && __has_builtin(__builtin_amdgcn_global_load_async_to_lds_b128)
#define HAVE_ASYNC_LDS 1
typedef int v4i_async __attribute__((ext_vector_type(4)));
#endif

// ---------------------------------------------------------------------------
// Kernel A: per-box preprocessing.
// Record layout (12 floats, 48B, 16B-aligned): c0x c0y c1x c1y c2x c2y c3x c3y
//                                              area meanx meany pad
// ---------------------------------------------------------------------------
__global__ __launch_bounds__(256) void diou_prep(
    const float* __restrict__ pred, const float* __restrict__ tgt,
    float* __restrict__ predD, float* __restrict__ tgtD, int NP, int NT)
{
  int t = blockIdx.x * blockDim.x + threadIdx.x;
  const float* box;
  float* outp;
  if (t < NP)            { box = pred + t * 5;        outp = predD + t * 12; }
  else if (t < NP + NT)  { box = tgt + (t - NP) * 5;  outp = tgtD + (t - NP) * 12; }
  else return;

  float cx = box[0], cy = box[1];
  float hx = box[2] * 0.5f, hy = box[3] * 0.5f;
  float ang = box[4];
  float ca = cosf(ang), sa = sinf(ang);

  const float ox[4] = {-1.f, -1.f, 1.f, 1.f};
  const float oy[4] = {-1.f,  1.f, 1.f, -1.f};
  float X[4], Y[4];
#pragma unroll
  for (int k = 0; k < 4; ++k) {
    float hxx = hx * ox[k];
    float hyy = hy * oy[k];
    X[k] = cx + hxx * ca - hyy * sa;
    Y[k] = cy + hxx * sa + hyy * ca;
  }
  // rect area via the reference's two-triangle cross formula
  float a0x = X[0] - X[2], a0y = Y[0] - Y[2];
  float b0x = X[1] - X[2], b0y = Y[1] - Y[2];
  float t1 = fabsf(a0x * b0y - a0y * b0x) * 0.5f;
  float a1x = X[0] - X[3], a1y = Y[0] - Y[3];
  float b1x = X[2] - X[3], b1y = Y[2] - Y[3];
  float t2 = fabsf(a1x * b1y - a1y * b1x) * 0.5f;

  float mx = (((X[0] + X[1]) + X[2]) + X[3]) * 0.25f;
  float my = (((Y[0] + Y[1]) + Y[2]) + Y[3]) * 0.25f;

  outp[0] = X[0]; outp[1] = Y[0];
  outp[2] = X[1]; outp[3] = Y[1];
  outp[4] = X[2]; outp[5] = Y[2];
  outp[6] = X[3]; outp[7] = Y[3];
  outp[8] = t1 + t2;
  outp[9] = mx; outp[10] = my; outp[11] = 0.f;
}

// ---------------------------------------------------------------------------
// Per-pair DIoU (faithful to reference semantics).
// ---------------------------------------------------------------------------
__device__ __forceinline__ float pair_diou(
    const float c1x[4], const float c1y[4], float area1, float m1x, float m1y,
    const float* __restrict__ tr)  // 12-float target record in LDS
{
  float c2x[4], c2y[4];
  c2x[0] = tr[0]; c2y[0] = tr[1];
  c2x[1] = tr[2]; c2y[1] = tr[3];
  c2x[2] = tr[4]; c2y[2] = tr[5];
  c2x[3] = tr[6]; c2y[3] = tr[7];
  float area = area1 + tr[8];
  float m2x = tr[9], m2y = tr[10];

  float qx[32], qy[32];
  bool pm[24];

  // corners of c1 inside rect c2  -> points 0..3
  {
    float ABx = c2x[1] - c2x[0], ABy = c2y[1] - c2y[0];
    float ADx = c2x[3] - c2x[0], ADy = c2y[3] - c2y[0];
    float dAB = ABx * ABx + ABy * ABy;
    float dAD = ADx * ADx + ADy * ADy;
#pragma unroll
    for (int k = 0; k < 4; ++k) {
      float APx = c1x[k] - c2x[0], APy = c1y[k] - c2y[0];
      float u = APx * ABx + APy * ABy;
      float v = APx * ADx + APy * ADy;
      qx[k] = c1x[k]; qy[k] = c1y[k];
      pm[k] = (u >= 0.f) && (u <= dAB) && (v >= 0.f) && (v <= dAD);
    }
  }
  // corners of c2 inside rect c1 -> points 4..7
  {
    float ABx = c1x[1] - c1x[0], ABy = c1y[1] - c1y[0];
    float ADx = c1x[3] - c1x[0], ADy = c1y[3] - c1y[0];
    float dAB = ABx * ABx + ABy * ABy;
    float dAD = ADx * ADx + ADy * ADy;
#pragma unroll
    for (int k = 0; k < 4; ++k) {
      float APx = c2x[k] - c1x[0], APy = c2y[k] - c1y[0];
      float u = APx * ABx + APy * ABy;
      float v = APx * ADx + APy * ADy;
      qx[4 + k] = c2x[k]; qy[4 + k] = c2y[k];
      pm[4 + k] = (u >= 0.f) && (u <= dAB) && (v >= 0.f) && (v <= dAD);
    }
  }
  // edge-edge intersections -> points 8..23 (index 8 + 4*i + j)
#pragma unroll
  for (int i = 0; i < 4; ++i) {
    float Ax = c1x[i], Ay = c1y[i];
    float Bx = c1x[(i + 1) & 3], By = c1y[(i + 1) & 3];
    float ABx = Bx - Ax, ABy = By - Ay;
    float cAB = Ax * By - Ay * Bx;
#pragma unroll
    for (int j = 0; j < 4; ++j) {
      float Cx = c2x[j], Cy = c2y[j];
      float Dx = c2x[(j + 1) & 3], Dy = c2y[(j + 1) & 3];
      float ACx = Cx - Ax, ACy = Cy - Ay;
      float ADx = Dx - Ax, ADy = Dy - Ay;
      float BCx = Cx - Bx, BCy = Cy - By;
      float BDx = Dx - Bx, BDy = Dy - By;
      bool s1 = ((ACx * ADy - ACy * ADx) > 0.f) != ((BCx * BDy - BCy * BDx) > 0.f);
      bool s2 = ((ABx * ACy - ABy * ACx) > 0.f) != ((ABx * ADy - ABy * ADx) > 0.f);
      bool valid = s1 && s2;
      float DCx = Dx - Cx, DCy = Dy - Cy;
      float den = DCx * ABy - DCy * ABx;
      float cCD = Cx * Dy - Cy * Dx;
      float d1 = (den == 0.f) ? 1.f : den;
      float ptx = (cAB * DCx - ABx * cCD) / d1;
      float pty = (cAB * DCy - ABy * cCD) / d1;
      int idx = 8 + 4 * i + j;
      qx[idx] = valid ? ptx : 0.f;
      qy[idx] = valid ? pty : 0.f;
      pm[idx] = valid;
    }
  }

  // masked centroid
  int cnt = 0;
  float sx = 0.f, sy = 0.f;
#pragma unroll
  for (int k = 0; k < 24; ++k) {
    cnt += pm[k] ? 1 : 0;
    sx += pm[k] ? qx[k] : 0.f;
    sy += pm[k] ? qy[k] : 0.f;
  }
  float denc = (float)(cnt > 0 ? cnt : 1);
  float ctrx = sx / denc, ctry = sy / denc;

  // center, mask -> (+inf,+inf) keys; pad to 32
#pragma unroll
  for (int k = 0; k < 24; ++k) {
    qx[k] = pm[k] ? (qx[k] - ctrx) : INFF;
    qy[k] = pm[k] ? (qy[k] - ctry) : INFF;
  }
#pragma unroll
  for (int k = 24; k < 32; ++k) { qx[k] = INFF; qy[k] = INFF; }

  // 32-wide bitonic sort, ascending lexicographic (x, then y).
  // Valid prefix ends sorted exactly as the reference's two stable argsorts;
  // the masked tail order is dead (triangles there are masked out).
#pragma unroll
  for (int k = 2; k <= 32; k <<= 1) {
#pragma unroll
    for (int j = k >> 1; j > 0; j >>= 1) {
#pragma unroll
      for (int i = 0; i < 32; ++i) {
        int l = i ^ j;
        if (l > i) {
          float ax = qx[i], ay = qy[i];
          float bx = qx[l], by = qy[l];
          bool gt = (ax > bx) || ((ax == bx) && (ay > by));
          bool lt = (ax < bx) || ((ax == bx) && (ay < by));
          bool sw = ((i & k) == 0) ? gt : lt;
          qx[i] = sw ? bx : ax;  qy[i] = sw ? by : ay;
          qx[l] = sw ? ax : bx;  qy[l] = sw ? ay : by;
        }
      }
    }
  }

  // triangle fan from p[0] over consecutive sorted points
  float p0x = qx[0] + ctrx, p0y = qy[0] + ctry;
  float inter = 0.f;
#pragma unroll
  for (int k = 0; k < 22; ++k) {
    float bx = qx[k + 1] + ctrx, by = qy[k + 1] + ctry;
    float cx = qx[k + 2] + ctrx, cy = qy[k + 2] + ctry;
    float ux = p0x - cx, uy = p0y - cy;
    float vx = bx - cx,  vy = by - cy;
    float tri = fabsf(ux * vy - uy * vx) * 0.5f;
    inter += ((k + 2) < cnt) ? tri : 0.f;
  }

  float iou = (cnt > 0) ? (inter / (area - inter + 1e-6f)) : 0.f;

  // max corner-to-corner distance (sqrt is monotone: max of squares first)
  float d2 = 0.f;
#pragma unroll
  for (int i = 0; i < 4; ++i) {
#pragma unroll
    for (int j = 0; j < 4; ++j) {
      float dx = c1x[i] - c2x[j];
      float dy = c1y[i] - c2y[j];
      d2 = fmaxf(d2, dx * dx + dy * dy);
    }
  }
  float d = sqrtf(d2);
  float cdx = m1x - m2x, cdy = m1y - m2y;
  float cd = sqrtf(cdx * cdx + cdy * cdy);
  float r = cd / (d + 1e-6f);
  return iou - r * r;
}

// ---------------------------------------------------------------------------
// Kernel B: one block per pred row; target records staged to LDS (async on
// gfx1250); streaming logsumexp across the 512 columns.
// ---------------------------------------------------------------------------
__global__ __launch_bounds__(256) void diou_rows(
    const float* __restrict__ predD, const float* __restrict__ tgtD,
    float* __restrict__ rowVals, int NT)
{
  __shared__ float sT[MAX_TGT * 12];
  __shared__ float sred[256];
  const int tid = threadIdx.x;
  const int row = blockIdx.x;

  const int total4 = (NT * 12) >> 2;  // float4 chunks (NT*12 divisible by 4)
#if defined(HAVE_ASYNC_LDS)
  for (int i = tid; i < total4; i += 256) {
    __builtin_amdgcn_global_load_async_to_lds_b128(
        (v4i_async*)(const_cast<float*>(tgtD) + i * 4),
        (v4i_async*)(sT + i * 4),
        0, 0);
  }
#if __has_builtin(__builtin_amdgcn_s_wait_asynccnt)
  __builtin_amdgcn_s_wait_asynccnt(0);
#else
  asm volatile("s_wait_asynccnt 0x0" ::: "memory");
#endif
#else
  for (int i = tid; i < total4; i += 256) {
    float4 v = ((const float4*)tgtD)[i];
    ((float4*)sT)[i] = v;
  }
#endif
  __syncthreads();

  const float* pr = predD + (size_t)row * 12;
  float c1x[4] = {pr[0], pr[2], pr[4], pr[6]};
  float c1y[4] = {pr[1], pr[3], pr[5], pr[7]};
  float area1 = pr[8], m1x = pr[9], m1y = pr[10];

  float mt = -INFF, st = 0.f, iou0 = 0.f;
#pragma clang loop unroll(disable)
  for (int col = tid; col < NT; col += 256) {
    float v = pair_diou(c1x, c1y, area1, m1x, m1y, sT + col * 12);
    if (col == 0) iou0 = v;  // only ever true for tid==0
    if (v > mt) { st = st * expf(mt - v) + 1.f; mt = v; }
    else        { st += expf(v - mt); }
  }

  // block max
  sred[tid] = mt; __syncthreads();
#pragma unroll
  for (int s = 128; s > 0; s >>= 1) {
    if (tid < s) sred[tid] = fmaxf(sred[tid], sred[tid + s]);
    __syncthreads();
  }
  float M = sred[0]; __syncthreads();

  // block sum of rescaled partial sums
  float contrib = (mt > -INFF) ? st * expf(mt - M) : 0.f;
  sred[tid] = contrib; __syncthreads();
#pragma unroll
  for (int s = 128; s > 0; s >>= 1) {
    if (tid < s) sred[tid] += sred[tid + s];
    __syncthreads();
  }
  if (tid == 0) rowVals[row] = M + logf(sred[0]) - iou0;
}

// ---------------------------------------------------------------------------
// Kernel C: mean over rows.
// ---------------------------------------------------------------------------
__global__ __launch_bounds__(256) void diou_reduce_rows(
    const float* __restrict__ rowVals, float* __restrict__ out, int NP)
{
  __shared__ float s[256];
  float a = 0.f;
  for (int i = threadIdx.x; i < NP; i += 256) a += rowVals[i];
  s[threadIdx.x] = a; __syncthreads();
#pragma unroll
  for (int st = 128; st > 0; st >>= 1) {
    if (threadIdx.x < st) s[threadIdx.x] += s[threadIdx.x + st];
    __syncthreads();
  }
  if (threadIdx.x == 0) out[0] = s[0] / (float)NP;
}

// ---------------------------------------------------------------------------
extern "C" void kernel_launch(void* const* d_in, const int* in_sizes, int n_in,
                              void* d_out, int out_size, void* d_ws, size_t ws_size,
                              hipStream_t stream) {
  const float* pred = (const float*)d_in[0];
  const float* tgt  = (const float*)d_in[1];
  const int NP = in_sizes[0] / 5;
  const int NT = in_sizes[1] / 5;

  float* ws = (float*)d_ws;
  float* predD   = ws;                         // NP*12 floats
  float* tgtD    = ws + (size_t)NP * 12;       // NT*12 floats
  float* rowVals = ws + (size_t)(NP + NT) * 12;// NP floats

  int tot = NP + NT;
  diou_prep<<<(tot + 255) / 256, 256, 0, stream>>>(pred, tgt, predD, tgtD, NP, NT);
  diou_rows<<<NP, 256, 0, stream>>>(predD, tgtD, rowVals, NT);
  diou_reduce_rows<<<1, 256, 0, stream>>>(rowVals, (float*)d_out, NP);
}